// PagedAttentionOp_24696061952306
// MI455X (gfx1250) — compile-verified
//
#include <hip/hip_runtime.h>

// Paged attention decode for MI455X (gfx1250, wave32).
// B=32, H=32, D=128, KVH=8 (G=4), P=2048, T=64, PPS=64.
// Memory-bound (~2 FLOP/byte): stream K/V once, skip pages beyond length,
// use V_WMMA_F32_16X16X4_F32 for QK^T and PV.

#define B_    32
#define H_    32
#define D_    128
#define KVH_  8
#define G_    4
#define P_    2048
#define T_    64
#define PPS_  64
#define NW    8          // waves per block (256 threads)

typedef __attribute__((ext_vector_type(2))) float v2f;
typedef __attribute__((ext_vector_type(8))) float v8f;

__device__ __forceinline__ v8f v8f_zero() {
    v8f z = {0.f, 0.f, 0.f, 0.f, 0.f, 0.f, 0.f, 0.f};
    return z;
}

__device__ __forceinline__ v8f wmma_f32(v2f a, v2f b, v8f c) {
    // V_WMMA_F32_16X16X4_F32:  D[16x16] = A[16x4] * B[4x16] + C
    return __builtin_amdgcn_wmma_f32_16x16x4_f32(
        /*neg_a=*/false, a, /*neg_b=*/false, b,
        /*c_mod=*/(short)0, c, /*reuse_a=*/false, /*reuse_b=*/false);
}

__global__ __launch_bounds__(NW * 32)
void paged_attn_f32_wmma(const float* __restrict__ q,
                         const float* __restrict__ k_pages,
                         const float* __restrict__ v_pages,
                         const int*   __restrict__ lengths,
                         const int*   __restrict__ page_indices,
                         float*       __restrict__ out)
{
    __shared__ float sm_s[NW][G_];
    __shared__ float sl_s[NW][G_];
    __shared__ float so_s[NW][G_][D_];   // 16 KB

    const int bk   = blockIdx.x;
    const int b    = bk / KVH_;
    const int kvh  = bk % KVH_;
    const int tid  = threadIdx.x;
    const int wid  = tid >> 5;
    const int lane = tid & 31;
    const int m    = lane & 15;   // column index (query within tile / d within tile)
    const int h    = lane >> 4;   // lane half

    const int len     = lengths[b];
    const int npages  = (len + T_ - 1) / T_;

    // ---- Q B-fragments for S^T = K_tile x Q^T, cached for the whole kernel ----
    // B-layout 4x16: VGPR0 -> k = d0 + 2h, VGPR1 -> k = d0 + 2h + 1; lane col = m.
    const float* qbase = q + ((size_t)b * H_ + (size_t)kvh * G_) * D_;
    v2f qf[32];
    #pragma unroll
    for (int c = 0; c < 32; ++c) {
        if (m < G_) {
            qf[c] = *(const v2f*)(qbase + (size_t)m * D_ + 4 * c + 2 * h);
        } else {
            v2f z; z.x = 0.f; z.y = 0.f;
            qf[c] = z;               // pad queries 4..15 with zeros
        }
    }

    // ---- flash-attention running state (per lane: query column m) ----
    float mstat = -1e30f;
    float lstat = 0.f;
    v8f acc[8];                      // O^T accumulators, 8 d-tiles of 16
    #pragma unroll
    for (int t = 0; t < 8; ++t) acc[t] = v8f_zero();

    const size_t head_off = (size_t)kvh * P_ * T_ * D_;
    const int*   ptab     = page_indices + (size_t)b * PPS_;

    for (int p = wid; p < npages; p += NW) {
        const int page = ptab[p];
        const float* kbase = k_pages + head_off + (size_t)page * T_ * D_;
        const float* vbase = v_pages + head_off + (size_t)page * T_ * D_;

        // prefetch next page for this wave (32 KB K + 32 KB V)
        const int pn = p + NW;
        if (pn < npages) {
            const int pg2 = ptab[pn];
            const float* kn = k_pages + head_off + (size_t)pg2 * T_ * D_;
            const float* vn = v_pages + head_off + (size_t)pg2 * T_ * D_;
            #pragma unroll
            for (int pf = 0; pf < 8; ++pf) {
                __builtin_prefetch(kn + (size_t)pf * 1024 + lane * 32, 0, 3);
                __builtin_prefetch(vn + (size_t)pf * 1024 + lane * 32, 0, 3);
            }
        }

        for (int tile = 0; tile < 4; ++tile) {
            const int tok0 = tile * 16;

            // ---- S^T[token][m] = sum_d K[token][d] * Q[m][d] ----
            // A-layout 16x4: lane row = token (tok0+m), VGPR0 -> d = 4c+2h, VGPR1 -> 4c+2h+1
            const float* krow = kbase + (size_t)(tok0 + m) * D_;
            v8f S = v8f_zero();
            #pragma unroll
            for (int c = 0; c < 32; ++c) {
                v2f a = *(const v2f*)(krow + 4 * c + 2 * h);
                S = wmma_f32(a, qf[c], S);
            }

            // ---- mask + online softmax update ----
            // C-layout: lane col = m (query), row token = v + 8h (local in tile)
            const int tb = p * T_ + tok0 + 8 * h;
            float tmax = -1e30f;
            float sv[8];
            #pragma unroll
            for (int v = 0; v < 8; ++v) {
                float s = ((tb + v) < len) ? S[v] : -1e30f;
                sv[v] = s;
                tmax  = fmaxf(tmax, s);
            }
            tmax = fmaxf(tmax, __shfl_xor(tmax, 16));
            const float mnew  = fmaxf(mstat, tmax);
            const float scale = __expf(mstat - mnew);
            float pe[8];
            float rsum = 0.f;
            #pragma unroll
            for (int v = 0; v < 8; ++v) {
                float e = (sv[v] > -1e29f) ? __expf(sv[v] - mnew) : 0.f;
                pe[v] = e;
                rsum += e;
            }
            rsum += __shfl_xor(rsum, 16);
            lstat = lstat * scale + rsum;
            mstat = mnew;
            #pragma unroll
            for (int t = 0; t < 8; ++t) {
                #pragma unroll
                for (int v = 0; v < 8; ++v) acc[t][v] *= scale;
            }

            // ---- O^T += V^T x P^T over 4 token-chunks of 4 ----
            #pragma unroll
            for (int c4 = 0; c4 < 4; ++c4) {
                // P^T token (tok0 + 4c4 + k) lives at vgpr (4c4+k)&7, lane half c4>>1.
                const int hs = c4 >> 1;
                const int j0 = (4 * c4) & 7;
                float p0 = pe[j0], p1 = pe[j0 + 1], p2 = pe[j0 + 2], p3 = pe[j0 + 3];
                float x0 = __shfl_xor(p0, 16);
                float x1 = __shfl_xor(p1, 16);
                float x2 = __shfl_xor(p2, 16);
                float x3 = __shfl_xor(p3, 16);
                // B-layout 4x16: VGPR0 -> token k = 2h, VGPR1 -> k = 2h+1 (within chunk)
                float b0, b1;
                if (h == hs) { b0 = h ? p2 : p0; b1 = h ? p3 : p1; }
                else         { b0 = h ? x2 : x0; b1 = h ? x3 : x1; }
                v2f bb; bb.x = b0; bb.y = b1;

                // A-layout 16x4: lane row = d (16t+m), VGPR0 -> token 2h, VGPR1 -> 2h+1
                const float* vrow = vbase + (size_t)(tok0 + 4 * c4 + 2 * h) * D_ + m;
                #pragma unroll
                for (int t = 0; t < 8; ++t) {
                    v2f a;
                    a.x = vrow[16 * t];
                    a.y = vrow[16 * t + D_];
                    acc[t] = wmma_f32(a, bb, acc[t]);
                }
            }
        }
    }

    // ---- per-wave results to LDS ----
    if (m < G_) {
        if (h == 0) { sm_s[wid][m] = mstat; sl_s[wid][m] = lstat; }
        #pragma unroll
        for (int t = 0; t < 8; ++t) {
            #pragma unroll
            for (int v = 0; v < 8; ++v) {
                so_s[wid][m][16 * t + v + 8 * h] = acc[t][v];
            }
        }
    }
    __syncthreads();

    // ---- cross-wave merge: 4 queries x 128 dims = 512 outputs, 2 per thread ----
    #pragma unroll
    for (int r = 0; r < 2; ++r) {
        const int idx = tid + r * 256;
        const int mm  = idx >> 7;      // 0..3
        const int dd  = idx & 127;
        float M = -1e30f;
        #pragma unroll
        for (int w = 0; w < NW; ++w) M = fmaxf(M, sm_s[w][mm]);
        float L = 0.f, val = 0.f;
        #pragma unroll
        for (int w = 0; w < NW; ++w) {
            const float sc = __expf(sm_s[w][mm] - M);
            L   += sl_s[w][mm] * sc;
            val += so_s[w][mm][dd] * sc;
        }
        out[((size_t)b * H_ + (size_t)(kvh * G_ + mm)) * D_ + dd] = val / L;
    }
}

extern "C" void kernel_launch(void* const* d_in, const int* in_sizes, int n_in,
                              void* d_out, int out_size, void* d_ws, size_t ws_size,
                              hipStream_t stream) {
    (void)in_sizes; (void)n_in; (void)out_size; (void)d_ws; (void)ws_size;
    const float* q            = (const float*)d_in[0];
    const float* k_pages      = (const float*)d_in[1];
    const float* v_pages      = (const float*)d_in[2];
    const int*   lengths      = (const int*)d_in[3];
    const int*   page_indices = (const int*)d_in[4];
    float*       out          = (float*)d_out;

    dim3 grid(B_ * KVH_);
    dim3 block(NW * 32);
    paged_attn_f32_wmma<<<grid, block, 0, stream>>>(q, k_pages, v_pages,
                                                    lengths, page_indices, out);
}